// GCN_6227702579850
// MI455X (gfx1250) — compile-verified
//
#include <hip/hip_runtime.h>

typedef float v2f __attribute__((ext_vector_type(2)));
typedef float v8f __attribute__((ext_vector_type(8)));

#define N_NODES 100000
#define E_EDGES 3200000
#define HID 256
#define OUTD 3

// ------------------------- degree / normalization -------------------------

__global__ void k_init_deg(float* deg, int n) {
    int i = blockIdx.x * blockDim.x + threadIdx.x;
    if (i < n) deg[i] = 1.0f;  // self-loop contribution
}

__global__ void k_deg_accum(const long long* __restrict__ dst, float* deg, int e) {
    int i = blockIdx.x * blockDim.x + threadIdx.x;
    if (i < e) atomicAdd(&deg[(int)dst[i]], 1.0f);
}

__global__ void k_dinv(float* deg, int n) {
    int i = blockIdx.x * blockDim.x + threadIdx.x;
    if (i < n) deg[i] = rsqrtf(deg[i]);  // deg >= 1 always (self loops)
}

// ------------------------- fp32 WMMA GEMM: H = X @ W -------------------------
// X: [nrows x 256], W: [256 x 256], H: [nrows x 256]
// Block: 256 threads = 8 waves. Tile: 128 rows x 64 cols, K chunked by 32.
// W tile is stored TRANSPOSED in LDS (Wt[n][k]) so each B fragment's (K,K+1)
// pair is one contiguous ds_load_b64 (no v_mov repacking). Next chunk's global
// loads are prefetched into registers during the WMMA phase.

#define TM 128
#define TN 64
#define TK 32
#define XS_LD 36   // padded: rows 16B-aligned, conflict-free A reads
#define WT_LD 34   // padded: 8B-aligned b64 reads, conflict-free across lanes

__global__ __launch_bounds__(256)
void k_gemm_wmma(const float* __restrict__ X, const float* __restrict__ W,
                 float* __restrict__ H, int nrows) {
    __shared__ float Xs[TM][XS_LD];  // 128 x 36 floats = 18.0 KB
    __shared__ float Wt[TN][WT_LD];  //  64 x 34 floats =  8.5 KB  (transposed)

    const int t    = threadIdx.x;
    const int wave = t >> 5;
    const int lane = t & 31;
    const int r0   = blockIdx.x * TM;
    const int n0   = blockIdx.y * TN;

    v8f acc0 = {}, acc1 = {}, acc2 = {}, acc3 = {};

    const int mrow = wave * 16;               // wave's row-tile inside block
    const int aRow = mrow + (lane & 15);      // M index this lane supplies
    const int kb   = (lane >> 4) << 1;        // lanes 0-15: K+0/1, lanes 16-31: K+2/3
    const int bn   = lane & 15;               // N index this lane supplies

    float4 xv[4], wv[2];

    auto load_global = [&](int k0) {
        #pragma unroll
        for (int i = 0; i < 4; ++i) {
            int idx = t + 256 * i;
            int row = idx >> 3;           // 0..127
            int kq  = idx & 7;
            xv[i] = make_float4(0.f, 0.f, 0.f, 0.f);
            if (r0 + row < nrows)
                xv[i] = *(const float4*)(X + (size_t)(r0 + row) * HID + k0 + kq * 4);
        }
        #pragma unroll
        for (int i = 0; i < 2; ++i) {
            int idx  = t + 256 * i;
            int krow = idx >> 4;          // 0..31
            int nq   = idx & 15;
            wv[i] = *(const float4*)(W + (size_t)(k0 + krow) * HID + n0 + nq * 4);
        }
    };

    auto store_lds = [&]() {
        #pragma unroll
        for (int i = 0; i < 4; ++i) {
            int idx = t + 256 * i;
            int row = idx >> 3;
            int kq  = idx & 7;
            *(float4*)&Xs[row][kq * 4] = xv[i];
        }
        #pragma unroll
        for (int i = 0; i < 2; ++i) {
            int idx  = t + 256 * i;
            int krow = idx >> 4;
            int nq   = idx & 15;
            Wt[nq * 4 + 0][krow] = wv[i].x;   // transpose on fill
            Wt[nq * 4 + 1][krow] = wv[i].y;
            Wt[nq * 4 + 2][krow] = wv[i].z;
            Wt[nq * 4 + 3][krow] = wv[i].w;
        }
    };

    load_global(0);
    store_lds();
    __syncthreads();

    for (int k0 = 0; k0 < HID; k0 += TK) {
        const bool more = (k0 + TK) < HID;
        if (more) load_global(k0 + TK);   // hide global latency behind WMMAs

        // ---- 8 K-steps of 4 => 32 v_wmma_f32_16x16x4_f32 per chunk ----
        #pragma unroll
        for (int kk = 0; kk < TK; kk += 4) {
            v2f a  = *(const v2f*)&Xs[aRow][kk + kb];
            v2f b0 = *(const v2f*)&Wt[bn +  0][kk + kb];
            v2f b1 = *(const v2f*)&Wt[bn + 16][kk + kb];
            v2f b2 = *(const v2f*)&Wt[bn + 32][kk + kb];
            v2f b3 = *(const v2f*)&Wt[bn + 48][kk + kb];
            acc0 = __builtin_amdgcn_wmma_f32_16x16x4_f32(false, a, false, b0, (short)0, acc0, false, false);
            acc1 = __builtin_amdgcn_wmma_f32_16x16x4_f32(false, a, false, b1, (short)0, acc1, false, false);
            acc2 = __builtin_amdgcn_wmma_f32_16x16x4_f32(false, a, false, b2, (short)0, acc2, false, false);
            acc3 = __builtin_amdgcn_wmma_f32_16x16x4_f32(false, a, false, b3, (short)0, acc3, false, false);
        }
        __syncthreads();
        if (more) {
            store_lds();
            __syncthreads();
        }
    }

    // ---- store D: VGPR r holds (M=r, N=lane0-15) / (M=r+8, N=lane-16) ----
    const int rbase = r0 + mrow + ((lane >> 4) << 3);
    const int col   = n0 + (lane & 15);
    #pragma unroll
    for (int r = 0; r < 8; ++r) {
        int row = rbase + r;
        if (row < nrows) {
            float* hp = H + (size_t)row * HID + col;
            hp[0]  = acc0[r];
            hp[16] = acc1[r];
            hp[32] = acc2[r];
            hp[48] = acc3[r];
        }
    }
}

// ------------------------- aggregation (SpMM), 256-wide -------------------------

__global__ void k_agg_init(const float* __restrict__ h, const float* __restrict__ dinv,
                           float* __restrict__ out, int total /* n*256 */) {
    int i = blockIdx.x * blockDim.x + threadIdx.x;
    if (i < total) {
        float w = dinv[i >> 8];
        out[i] = h[i] * w * w;  // self-loop term
    }
}

__global__ void k_agg_edges(const float* __restrict__ h,
                            const long long* __restrict__ src,
                            const long long* __restrict__ dst,
                            const float* __restrict__ dinv,
                            float* __restrict__ out, int e) {
    int ed = blockIdx.x * (blockDim.x >> 5) + (threadIdx.x >> 5);  // wave per edge
    if (ed >= e) return;
    int lane = threadIdx.x & 31;
    int s = (int)src[ed];
    int d = (int)dst[ed];
    float w = dinv[s] * dinv[d];
    const float4* hv = (const float4*)(h + (size_t)s * HID + lane * 8);  // 2 x b128 gather
    float*        od = out + (size_t)d * HID + lane * 8;
    float4 v0 = hv[0];
    float4 v1 = hv[1];
    atomicAdd(&od[0], v0.x * w);
    atomicAdd(&od[1], v0.y * w);
    atomicAdd(&od[2], v0.z * w);
    atomicAdd(&od[3], v0.w * w);
    atomicAdd(&od[4], v1.x * w);
    atomicAdd(&od[5], v1.y * w);
    atomicAdd(&od[6], v1.z * w);
    atomicAdd(&od[7], v1.w * w);
}

__global__ void k_bias_relu(float* __restrict__ out, const float* __restrict__ b,
                            int total) {
    int i = blockIdx.x * blockDim.x + threadIdx.x;
    if (i < total) out[i] = fmaxf(out[i] + b[i & 255], 0.0f);
}

// ------------------------- layer 3: skinny GEMM + 3-wide aggregation -------------------------

__global__ void k_out3(const float* __restrict__ A, const float* __restrict__ W3,
                       float* __restrict__ H3, int n) {
    int i = blockIdx.x * blockDim.x + threadIdx.x;
    if (i >= n) return;
    const float4* a = (const float4*)(A + (size_t)i * HID);
    float s0 = 0.f, s1 = 0.f, s2 = 0.f;
    #pragma unroll 8
    for (int q = 0; q < HID / 4; ++q) {
        float4 v = a[q];
        const float* w = W3 + q * 12;  // 4 K-rows x 3 cols
        s0 = fmaf(v.x, w[0], s0);  s1 = fmaf(v.x, w[1],  s1);  s2 = fmaf(v.x, w[2],  s2);
        s0 = fmaf(v.y, w[3], s0);  s1 = fmaf(v.y, w[4],  s1);  s2 = fmaf(v.y, w[5],  s2);
        s0 = fmaf(v.z, w[6], s0);  s1 = fmaf(v.z, w[7],  s1);  s2 = fmaf(v.z, w[8],  s2);
        s0 = fmaf(v.w, w[9], s0);  s1 = fmaf(v.w, w[10], s1);  s2 = fmaf(v.w, w[11], s2);
    }
    H3[(size_t)i * 3 + 0] = s0;
    H3[(size_t)i * 3 + 1] = s1;
    H3[(size_t)i * 3 + 2] = s2;
}

__global__ void k_agg3_init(const float* __restrict__ h3, const float* __restrict__ dinv,
                            float* __restrict__ out, int n) {
    int i = blockIdx.x * blockDim.x + threadIdx.x;
    if (i < n) {
        float w = dinv[i] * dinv[i];
        out[i * 3 + 0] = h3[i * 3 + 0] * w;
        out[i * 3 + 1] = h3[i * 3 + 1] * w;
        out[i * 3 + 2] = h3[i * 3 + 2] * w;
    }
}

__global__ void k_agg3_edges(const float* __restrict__ h3,
                             const long long* __restrict__ src,
                             const long long* __restrict__ dst,
                             const float* __restrict__ dinv,
                             float* __restrict__ out, int e) {
    int i = blockIdx.x * blockDim.x + threadIdx.x;
    if (i >= e) return;
    int s = (int)src[i];
    int d = (int)dst[i];
    float w = dinv[s] * dinv[d];
    atomicAdd(&out[d * 3 + 0], h3[s * 3 + 0] * w);
    atomicAdd(&out[d * 3 + 1], h3[s * 3 + 1] * w);
    atomicAdd(&out[d * 3 + 2], h3[s * 3 + 2] * w);
}

__global__ void k_bias3(float* __restrict__ out, const float* __restrict__ b3, int n) {
    int i = blockIdx.x * blockDim.x + threadIdx.x;
    if (i < n) {
        out[i * 3 + 0] += b3[0];
        out[i * 3 + 1] += b3[1];
        out[i * 3 + 2] += b3[2];
    }
}

// ------------------------- launcher -------------------------

extern "C" void kernel_launch(void* const* d_in, const int* in_sizes, int n_in,
                              void* d_out, int out_size, void* d_ws, size_t ws_size,
                              hipStream_t stream) {
    const float*     x    = (const float*)d_in[0];
    const long long* eidx = (const long long*)d_in[1];  // int64 in reference
    const float*     W1   = (const float*)d_in[2];
    const float*     b1   = (const float*)d_in[3];
    const float*     W2   = (const float*)d_in[4];
    const float*     b2   = (const float*)d_in[5];
    const float*     W3   = (const float*)d_in[6];
    const float*     b3   = (const float*)d_in[7];
    float*           out  = (float*)d_out;

    const long long* src = eidx;            // edge_idx[0]
    const long long* dst = eidx + E_EDGES;  // edge_idx[1]

    float* ws   = (float*)d_ws;
    float* dinv = ws;                                   // N floats
    float* bufH = ws + 100352;                          // N x 256 (padded offset, 16B aligned)
    float* bufG = bufH + (size_t)N_NODES * HID;         // N x 256
    float* h3   = bufH;                                 // reuse: N x 3 (bufH dead by then)

    const int TPB = 256;
    const int nTot = N_NODES * HID;
    dim3 ggrid((N_NODES + TM - 1) / TM, HID / TN);

    // normalization (reused across layers)
    k_init_deg <<<(N_NODES + TPB - 1) / TPB, TPB, 0, stream>>>(dinv, N_NODES);
    k_deg_accum<<<(E_EDGES + TPB - 1) / TPB, TPB, 0, stream>>>(dst, dinv, E_EDGES);
    k_dinv     <<<(N_NODES + TPB - 1) / TPB, TPB, 0, stream>>>(dinv, N_NODES);

    // layer 1: H = X@W1 ; A = Agg(H) + b1 ; relu
    k_gemm_wmma<<<ggrid, TPB, 0, stream>>>(x, W1, bufH, N_NODES);
    k_agg_init <<<(nTot + TPB - 1) / TPB, TPB, 0, stream>>>(bufH, dinv, bufG, nTot);
    k_agg_edges<<<(E_EDGES + 7) / 8, TPB, 0, stream>>>(bufH, src, dst, dinv, bufG, E_EDGES);
    k_bias_relu<<<(nTot + TPB - 1) / TPB, TPB, 0, stream>>>(bufG, b1, nTot);

    // layer 2: H = A@W2 ; A = Agg(H) + b2 ; relu
    k_gemm_wmma<<<ggrid, TPB, 0, stream>>>(bufG, W2, bufH, N_NODES);
    k_agg_init <<<(nTot + TPB - 1) / TPB, TPB, 0, stream>>>(bufH, dinv, bufG, nTot);
    k_agg_edges<<<(E_EDGES + 7) / 8, TPB, 0, stream>>>(bufH, src, dst, dinv, bufG, E_EDGES);
    k_bias_relu<<<(nTot + TPB - 1) / TPB, TPB, 0, stream>>>(bufG, b2, nTot);

    // layer 3: H3 = A@W3 ; out = Agg(H3) + b3
    k_out3      <<<(N_NODES + TPB - 1) / TPB, TPB, 0, stream>>>(bufG, W3, h3, N_NODES);
    k_agg3_init <<<(N_NODES + TPB - 1) / TPB, TPB, 0, stream>>>(h3, dinv, out, N_NODES);
    k_agg3_edges<<<(E_EDGES + TPB - 1) / TPB, TPB, 0, stream>>>(h3, src, dst, dinv, out, E_EDGES);
    k_bias3     <<<(N_NODES + TPB - 1) / TPB, TPB, 0, stream>>>(out, b3, N_NODES);
}